// GraphMF_25305947308735
// MI455X (gfx1250) — compile-verified
//
#include <hip/hip_runtime.h>
#include <math.h>

#define DIM   128

typedef __attribute__((ext_vector_type(2))) float v2f;
typedef __attribute__((ext_vector_type(8))) float v8f;

// ---------------------------------------------------------------- utilities

__global__ __launch_bounds__(256) void zero_kernel(float* __restrict__ p, long n4) {
  long i = (long)blockIdx.x * blockDim.x + threadIdx.x;
  if (i < n4) reinterpret_cast<float4*>(p)[i] = make_float4(0.f, 0.f, 0.f, 0.f);
}

__global__ __launch_bounds__(256) void scatter_deg_kernel(const int* __restrict__ dst,
                                                          float* __restrict__ deg, int E) {
  int e = blockIdx.x * blockDim.x + threadIdx.x;
  if (e < E)
    __hip_atomic_fetch_add(&deg[dst[e]], 1.0f, __ATOMIC_RELAXED, __HIP_MEMORY_SCOPE_AGENT);
}

__global__ __launch_bounds__(256) void inv_kernel(float* __restrict__ deg, int n) {
  int i = blockIdx.x * blockDim.x + threadIdx.x;
  if (i < n) deg[i] = 1.0f / (deg[i] + 1.0f);
}

// One wave per edge; each lane moves 4 contiguous floats with hardware f32 atomics.
__global__ __launch_bounds__(256) void scatter_feats_kernel(const float* __restrict__ feats,
                                                            const int* __restrict__ src,
                                                            const int* __restrict__ dst,
                                                            float* __restrict__ neigh, int E) {
  int t = blockIdx.x * blockDim.x + threadIdx.x;
  int e = t >> 5;
  int lane = t & 31;
  if (e >= E) return;
  int s = src[e], d = dst[e];
  float4 v = reinterpret_cast<const float4*>(feats + (size_t)s * DIM)[lane];
  float* base = neigh + (size_t)d * DIM + lane * 4;
  __hip_atomic_fetch_add(base + 0, v.x, __ATOMIC_RELAXED, __HIP_MEMORY_SCOPE_AGENT);
  __hip_atomic_fetch_add(base + 1, v.y, __ATOMIC_RELAXED, __HIP_MEMORY_SCOPE_AGENT);
  __hip_atomic_fetch_add(base + 2, v.z, __ATOMIC_RELAXED, __HIP_MEMORY_SCOPE_AGENT);
  __hip_atomic_fetch_add(base + 3, v.w, __ATOMIC_RELAXED, __HIP_MEMORY_SCOPE_AGENT);
}

// ---------------------------------------------------------------- WMMA GEMMs
// out[16 x 128] tile per block; 8 waves, each wave one 16-wide N-tile.
// A tile staged in LDS; f32 WMMA 16x16x4, K-loop over DIM.
// A-frag (v2f): lane L (half=L>>4, m=L&15): a = A[m][4k+2*half + {0,1}]
// B-frag (v2f): col = n0 + (L&15):          b = W[4k+2*half + {0,1}][col]
// C/D (v8f):    elem r -> row = r + 8*half, col = n0 + (L&15)

__global__ __launch_bounds__(256) void sage_gemm_kernel(const float* __restrict__ feats,
                                                        const float* __restrict__ neigh,
                                                        const float* __restrict__ inv,
                                                        const float* __restrict__ W,
                                                        const float* __restrict__ bias,
                                                        float* __restrict__ out) {
  __shared__ float As[16 * DIM];
  int r0 = blockIdx.x * 16;
  for (int i = threadIdx.x; i < 16 * DIM; i += 256) {
    int rr = i >> 7, cc = i & 127;
    int node = r0 + rr;
    As[i] = (neigh[(size_t)node * DIM + cc] + feats[(size_t)node * DIM + cc]) * inv[node];
  }
  __syncthreads();  // after this, all global reads of `neigh` rows are done -> out may alias neigh

  int lane = threadIdx.x & 31;
  int wave = threadIdx.x >> 5;
  int half = lane >> 4;
  int l16  = lane & 15;
  int n0 = wave * 16;
  v8f acc = {};
#pragma unroll 4
  for (int k4 = 0; k4 < DIM / 4; ++k4) {
    int kb = k4 * 4 + half * 2;
    v2f a, b;
    a.x = As[l16 * DIM + kb];
    a.y = As[l16 * DIM + kb + 1];
    b.x = W[(size_t)kb * DIM + n0 + l16];
    b.y = W[(size_t)(kb + 1) * DIM + n0 + l16];
    acc = __builtin_amdgcn_wmma_f32_16x16x4_f32(false, a, false, b, (short)0, acc, false, false);
  }
  int col = n0 + l16;
  float bc = bias[col];
#pragma unroll
  for (int r = 0; r < 8; ++r) {
    int row = r0 + r + half * 8;
    out[(size_t)row * DIM + col] = acc[r] + bc;
  }
}

template <int K>
__global__ __launch_bounds__(256) void mlp_gemm_kernel(const float* __restrict__ x,
                                                       const float* __restrict__ W,
                                                       const float* __restrict__ bias,
                                                       float* __restrict__ out) {
  __shared__ float As[16 * K];
  int r0 = blockIdx.x * 16;
  for (int i = threadIdx.x; i < 16 * K; i += 256) {
    int rr = i / K, cc = i % K;
    As[i] = x[(size_t)(r0 + rr) * K + cc];
  }
  __syncthreads();

  int lane = threadIdx.x & 31;
  int wave = threadIdx.x >> 5;
  int half = lane >> 4;
  int l16  = lane & 15;
  int n0 = wave * 16;
  v8f acc = {};
#pragma unroll 4
  for (int k4 = 0; k4 < K / 4; ++k4) {
    int kb = k4 * 4 + half * 2;
    v2f a, b;
    a.x = As[l16 * K + kb];
    a.y = As[l16 * K + kb + 1];
    b.x = W[(size_t)kb * DIM + n0 + l16];
    b.y = W[(size_t)(kb + 1) * DIM + n0 + l16];
    acc = __builtin_amdgcn_wmma_f32_16x16x4_f32(false, a, false, b, (short)0, acc, false, false);
  }
  int col = n0 + l16;
  float bc = bias[col];
#pragma unroll
  for (int r = 0; r < 8; ++r) {
    int row = r0 + r + half * 8;
    out[(size_t)row * DIM + col] = acc[r] + bc;
  }
}

// ------------------------------------------------- LayerNorm + activation
// One wave per 128-wide row; 8 rows per 256-thread block. act: 0 = ELU, 1 = ReLU.
__device__ __forceinline__ float elu_f(float x) { return x > 0.f ? x : expm1f(x); }

__global__ __launch_bounds__(256) void ln_act_kernel(const float* __restrict__ in,
                                                     float* __restrict__ out,
                                                     const float* __restrict__ g,
                                                     const float* __restrict__ be,
                                                     int M, int act) {
  int lane = threadIdx.x & 31;
  int row = blockIdx.x * 8 + (threadIdx.x >> 5);
  if (row >= M) return;
  float4 v = reinterpret_cast<const float4*>(in + (size_t)row * DIM)[lane];
  float s1 = v.x + v.y + v.z + v.w;
  float s2 = v.x * v.x + v.y * v.y + v.z * v.z + v.w * v.w;
#pragma unroll
  for (int m = 16; m > 0; m >>= 1) {
    s1 += __shfl_xor(s1, m, 32);
    s2 += __shfl_xor(s2, m, 32);
  }
  float mu = s1 * (1.0f / DIM);
  float var = s2 * (1.0f / DIM) - mu * mu;
  float rs = rsqrtf(var + 1e-5f);
  float4 gg = reinterpret_cast<const float4*>(g)[lane];
  float4 bb = reinterpret_cast<const float4*>(be)[lane];
  float4 o;
  o.x = (v.x - mu) * rs * gg.x + bb.x;
  o.y = (v.y - mu) * rs * gg.y + bb.y;
  o.z = (v.z - mu) * rs * gg.z + bb.z;
  o.w = (v.w - mu) * rs * gg.w + bb.w;
  if (act == 0) { o.x = elu_f(o.x); o.y = elu_f(o.y); o.z = elu_f(o.z); o.w = elu_f(o.w); }
  else          { o.x = fmaxf(o.x, 0.f); o.y = fmaxf(o.y, 0.f); o.z = fmaxf(o.z, 0.f); o.w = fmaxf(o.w, 0.f); }
  reinterpret_cast<float4*>(out + (size_t)row * DIM)[lane] = o;
}

// ------------------------------------------------- batch gather + MLP head

__global__ __launch_bounds__(256) void gather_kernel(const float* __restrict__ uf,
                                                     const float* __restrict__ sf,
                                                     const int* __restrict__ uIdx,
                                                     const int* __restrict__ iIdx,
                                                     float* __restrict__ x, int B) {
  int t = blockIdx.x * blockDim.x + threadIdx.x;
  int b = t >> 6;          // 64 float4 chunks per 256-wide row
  int q = t & 63;
  if (b >= B) return;
  float4 v;
  if (q < 32) v = reinterpret_cast<const float4*>(uf + (size_t)uIdx[b] * DIM)[q];
  else        v = reinterpret_cast<const float4*>(sf + (size_t)iIdx[b] * DIM)[q - 32];
  reinterpret_cast<float4*>(x + (size_t)b * 2 * DIM)[q] = v;
}

__global__ __launch_bounds__(256) void head_kernel(const float* __restrict__ h,
                                                   const float* __restrict__ W3,
                                                   const float* __restrict__ b3,
                                                   float* __restrict__ out, int B) {
  int b = blockIdx.x * blockDim.x + threadIdx.x;
  if (b >= B) return;
  const float* row = h + (size_t)b * DIM;
  float acc = 0.f;
#pragma unroll 8
  for (int k = 0; k < DIM; ++k) acc += row[k] * W3[k];
  float z = acc + b3[0];
  out[b] = 1.0f / (1.0f + expf(-z));
}

// ---------------------------------------------------------------- launcher

extern "C" void kernel_launch(void* const* d_in, const int* in_sizes, int n_in,
                              void* d_out, int out_size, void* d_ws, size_t ws_size,
                              hipStream_t stream) {
  const int*   userIdx  = (const int*)d_in[0];
  const int*   itemIdx  = (const int*)d_in[1];
  const int*   user_src = (const int*)d_in[2];
  const int*   user_dst = (const int*)d_in[3];
  const int*   serv_src = (const int*)d_in[4];
  const int*   serv_dst = (const int*)d_in[5];
  const float* user_emb = (const float*)d_in[6];
  const float* serv_emb = (const float*)d_in[7];
  const float* Wu  = (const float*)d_in[8];
  const float* bu  = (const float*)d_in[9];
  const float* gu  = (const float*)d_in[10];
  const float* beu = (const float*)d_in[11];
  const float* Ws  = (const float*)d_in[12];
  const float* bs  = (const float*)d_in[13];
  const float* gs  = (const float*)d_in[14];
  const float* bes = (const float*)d_in[15];
  const float* W1  = (const float*)d_in[16];
  const float* b1  = (const float*)d_in[17];
  const float* g1  = (const float*)d_in[18];
  const float* be1 = (const float*)d_in[19];
  const float* W2  = (const float*)d_in[20];
  const float* b2  = (const float*)d_in[21];
  const float* g2  = (const float*)d_in[22];
  const float* be2 = (const float*)d_in[23];
  const float* W3  = (const float*)d_in[24];
  const float* b3  = (const float*)d_in[25];

  const int B = in_sizes[0];                 // 16384
  const int E = in_sizes[2];                 // 800000
  const int N = in_sizes[6] / DIM;           // 50000
  const size_t NF = (size_t)N * DIM;

  float* f     = (float*)d_ws;
  float* neigh = f;                          // NF   (shared scratch, reused as GEMM out)
  float* ufA   = neigh + NF;                 // NF
  float* ufB   = ufA + NF;                   // NF   (final user feats)
  float* sfA   = ufB + NF;                   // NF
  float* sfB   = sfA + NF;                   // NF   (final serv feats)
  float* uinv  = sfB + NF;                   // N
  float* sinv  = uinv + N;                   // N
  float* x  = ufA;                           // [B, 256]  (ufA free after stacks)
  float* h1 = sfA;                           // [B, 128]  (sfA free after stacks)
  float* h2 = sfA + (size_t)B * DIM;         // [B, 128]

  dim3 blk(256);
  int gZeroN  = (int)((N + 1023) / 1024);              // zero N floats (4/thread)
  int gZeroNF = (int)((NF + 1023) / 1024);
  int gE      = (E + 255) / 256;
  int gN      = (N + 255) / 256;
  int gEdge32 = (int)(((size_t)E * 32 + 255) / 256);
  int gTiles  = N / 16;                                // 3125 (N % 16 == 0)
  int gLN     = (N + 7) / 8;
  int gBT     = B / 16;                                // 1024
  int gBLN    = (B + 7) / 8;

  // ---- degrees -> inv = 1/(deg+1)
  zero_kernel<<<gZeroN, blk, 0, stream>>>(uinv, N / 4);
  zero_kernel<<<gZeroN, blk, 0, stream>>>(sinv, N / 4);
  scatter_deg_kernel<<<gE, blk, 0, stream>>>(user_dst, uinv, E);
  scatter_deg_kernel<<<gE, blk, 0, stream>>>(serv_dst, sinv, E);
  inv_kernel<<<gN, blk, 0, stream>>>(uinv, N);
  inv_kernel<<<gN, blk, 0, stream>>>(sinv, N);

  // ---- user SAGE stack: emb -> ufB -> ufA -> ufB
  {
    const float* fin = user_emb;
    float* outs[3] = {ufB, ufA, ufB};
    for (int l = 0; l < 3; ++l) {
      zero_kernel<<<gZeroNF, blk, 0, stream>>>(neigh, (long)(NF / 4));
      scatter_feats_kernel<<<gEdge32, blk, 0, stream>>>(fin, user_src, user_dst, neigh, E);
      sage_gemm_kernel<<<gTiles, blk, 0, stream>>>(fin, neigh, uinv,
                                                   Wu + (size_t)l * DIM * DIM, bu + l * DIM, neigh);
      ln_act_kernel<<<gLN, blk, 0, stream>>>(neigh, outs[l], gu + l * DIM, beu + l * DIM, N, 0);
      fin = outs[l];
    }
  }
  // ---- service SAGE stack: emb -> sfB -> sfA -> sfB
  {
    const float* fin = serv_emb;
    float* outs[3] = {sfB, sfA, sfB};
    for (int l = 0; l < 3; ++l) {
      zero_kernel<<<gZeroNF, blk, 0, stream>>>(neigh, (long)(NF / 4));
      scatter_feats_kernel<<<gEdge32, blk, 0, stream>>>(fin, serv_src, serv_dst, neigh, E);
      sage_gemm_kernel<<<gTiles, blk, 0, stream>>>(fin, neigh, sinv,
                                                   Ws + (size_t)l * DIM * DIM, bs + l * DIM, neigh);
      ln_act_kernel<<<gLN, blk, 0, stream>>>(neigh, outs[l], gs + l * DIM, bes + l * DIM, N, 0);
      fin = outs[l];
    }
  }

  // ---- MLP head
  gather_kernel<<<(int)(((size_t)B * 64 + 255) / 256), blk, 0, stream>>>(ufB, sfB, userIdx, itemIdx, x, B);
  mlp_gemm_kernel<2 * DIM><<<gBT, blk, 0, stream>>>(x, W1, b1, h1);
  ln_act_kernel<<<gBLN, blk, 0, stream>>>(h1, h1, g1, be1, B, 1);
  mlp_gemm_kernel<DIM><<<gBT, blk, 0, stream>>>(h1, W2, b2, h2);
  ln_act_kernel<<<gBLN, blk, 0, stream>>>(h2, h2, g2, be2, B, 1);
  head_kernel<<<(B + 255) / 256, blk, 0, stream>>>(h2, W3, b3, (float*)d_out, B);
}